// FaceNurbsLoss_21560735826215
// MI455X (gfx1250) — compile-verified
//
#include <hip/hip_runtime.h>
#include <math.h>

// ---------------------------------------------------------------------------
// FaceNurbsLoss for MI455X (gfx1250, wave32).
//
// Key algebraic reduction: surf[b,u,v,c] = curve[b,u,c]*sv[v] means every
// 3072-dim sample row is K * vec(curve_b) with K having orthogonal equal-norm
// columns (norm = ||sv||). Hence the 3072-dim PCA reduces exactly to a 96-dim
// PCA of the curve vectors, and dist_pca = ||sv|| * ||W3^T (mean_p - mean_l)||.
// The IoU term reduces to counting sv[v]*||dcurve[b,u]|| < 0.5.
//
// GEMMs use V_WMMA_F32_16X16X4_F32 (native f32 WMMA, no precision loss):
//   curve:  M=2048(b) x N=32(u) x K=68(i), per channel c and per tensor.
//   cov:    M=96 x N=96 x K=4096 (centered combined curve rows).
// ---------------------------------------------------------------------------

#define NJ     68
#define RES    32
#define BATCH  2048
#define XCOLS  96            // 32 u-samples * 3 channels
#define NROWS  4096          // combined pred+label rows for PCA

typedef __attribute__((ext_vector_type(2))) float v2f;
typedef __attribute__((ext_vector_type(8))) float v8f;

// Workspace layout (float offsets). Total ~404848 floats (~1.6 MB).
#define WS_NU    0           // 68*32 basis (p=2)
#define WS_SV    2176        // 32: column sums of q=3 basis
#define WS_CNT   2208        // 1 uint counter (IoU inter)
#define WS_MEANP 2224        // 96
#define WS_MEANL 2320        // 96
#define WS_COV   2416        // 96*96
#define WS_XP    11632       // 2048*96 curve(pred), row = [u*3+c]
#define WS_XL    208240      // 2048*96 curve(label)

// ----------------------------------------------------------------- basis ---
__device__ void eval_basis(int n, int p, float u, float* Nout) {
  const int L = n + p + 1;              // <= 72
  float kn[80];
  float N[80];
  const float denom = (float)(L - 2 * p - 1);
  for (int j = 0; j < L; ++j) {
    float v;
    if (j <= p)              v = 0.f;
    else if (j >= L - p - 1) v = 1.f;
    else                     v = (float)(j - p) / denom;
    kn[j] = v;
  }
  int rows = L - 1;
  for (int i = 0; i < rows; ++i)
    N[i] = (kn[i] <= u && u < kn[i + 1]) ? 1.f : 0.f;   // half-open, as ref
  for (int d = 1; d <= p; ++d) {
    int m = rows - 1;
    for (int i = 0; i < m; ++i) {       // in-place front-to-back is safe
      float ld = kn[i + d] - kn[i];
      float rd = kn[i + d + 1] - kn[i + 1];
      float t1 = (ld == 0.f) ? 0.f : (u - kn[i]) / ld * N[i];
      float t2 = (rd == 0.f) ? 0.f : (kn[i + d + 1] - u) / rd * N[i + 1];
      N[i] = t1 + t2;
    }
    rows = m;
  }
  for (int i = 0; i < n; ++i) Nout[i] = N[i];
}

__global__ void fnl_basis_kernel(float* ws) {
  int u_idx = threadIdx.x;
  if (u_idx == 0) *((unsigned int*)(ws + WS_CNT)) = 0u;   // zero IoU counter
  if (u_idx >= RES) return;
  float u = (float)u_idx / (float)(RES - 1);
  float Nb[NJ];
  eval_basis(NJ, 2, u, Nb);
  for (int i = 0; i < NJ; ++i) ws[WS_NU + i * RES + u_idx] = Nb[i];
  eval_basis(NJ, 3, u, Nb);
  float s = 0.f;
  for (int i = 0; i < NJ; ++i) s += Nb[i];
  ws[WS_SV + u_idx] = s;
}

// -------------------------------------------------- curve GEMMs via WMMA ---
// curve[b,u,c] = sum_i cp[b,i,c] * Nu[i,u]; 1536 wave-jobs, one 16x16 D tile
// each (128 m-tiles x 2 n-tiles x 3 channels x 2 tensors), K=68 = 17 steps.
__global__ void fnl_curve_wmma_kernel(const float* __restrict__ pred,
                                      const float* __restrict__ label,
                                      float* __restrict__ ws) {
  const float* Nu = ws + WS_NU;
  int wave = (blockIdx.x * blockDim.x + threadIdx.x) >> 5;
  int lane = threadIdx.x & 31;
  int mt = wave & 127;
  int rest = wave >> 7;
  int nt = rest & 1;  rest >>= 1;
  int c  = rest % 3;
  int t  = rest / 3;
  const float* src = t ? label : pred;
  float* X = ws + (t ? WS_XL : WS_XP);
  int b0 = mt * 16, u0 = nt * 16;
  int m    = lane & 15;
  int koff = (lane >> 4) << 1;          // A/B 16x4 f32 layout: K = k0+koff+{0,1}
  v8f acc = {0.f, 0.f, 0.f, 0.f, 0.f, 0.f, 0.f, 0.f};
  for (int k0 = 0; k0 < NJ; k0 += 4) {
    int kb = k0 + koff;
    v2f av, bv;
    av.x = src[((b0 + m) * NJ + kb    ) * 3 + c];
    av.y = src[((b0 + m) * NJ + kb + 1) * 3 + c];
    bv.x = Nu[(kb    ) * RES + u0 + m];
    bv.y = Nu[(kb + 1) * RES + u0 + m];
    acc = __builtin_amdgcn_wmma_f32_16x16x4_f32(false, av, false, bv,
                                                (short)0, acc, false, false);
  }
  int u = u0 + m;
  int mhi = (lane >> 4) << 3;           // C/D: VGPR r -> M = r + 8*(lane/16)
#pragma unroll
  for (int r = 0; r < 8; ++r) {
    int row = b0 + r + mhi;
    X[row * XCOLS + u * 3 + c] = acc[r];
  }
}

// ------------------------------------------------------------ column means --
__global__ void fnl_means_kernel(float* ws) {
  int tid = threadIdx.x;
  if (tid >= 2 * XCOLS) return;
  int j = tid % XCOLS;
  int t = tid / XCOLS;
  const float* X = ws + (t ? WS_XL : WS_XP);
  float s = 0.f;
  for (int r = 0; r < BATCH; ++r) s += X[r * XCOLS + j];
  ws[(t ? WS_MEANL : WS_MEANP) + j] = s / (float)BATCH;
}

// ---------------------------------------- 96x96 covariance GEMM via WMMA ---
// Cov = Xc^T Xc / 4095, Xc = centered combined rows (4096 x 96).
// 36 blocks (6x6 tiles) x 8 waves; each wave takes a 512-row K slice,
// partial tiles reduced through LDS (single deterministic writer).
__global__ void fnl_cov_wmma_kernel(float* __restrict__ ws) {
  __shared__ float cm[XCOLS];
  __shared__ float red[8 * 8 * 32];
  int tid = threadIdx.x;
  if (tid < XCOLS) cm[tid] = 0.5f * (ws[WS_MEANP + tid] + ws[WS_MEANL + tid]);
  __syncthreads();
  const float* Xp = ws + WS_XP;
  const float* Xl = ws + WS_XL;
  int tile = blockIdx.x;
  int i0 = (tile / 6) * 16, j0 = (tile % 6) * 16;
  int w = tid >> 5, lane = tid & 31;
  int m = lane & 15;
  int koff = (lane >> 4) << 1;
  v8f acc = {0.f, 0.f, 0.f, 0.f, 0.f, 0.f, 0.f, 0.f};
  int kbeg = w * 512;
  for (int k0 = kbeg; k0 < kbeg + 512; k0 += 4) {
    int r0 = k0 + koff;
    int r1 = r0 + 1;
    const float* row0 = (r0 < BATCH) ? (Xp + r0 * XCOLS) : (Xl + (r0 - BATCH) * XCOLS);
    const float* row1 = (r1 < BATCH) ? (Xp + r1 * XCOLS) : (Xl + (r1 - BATCH) * XCOLS);
    v2f av, bv;
    av.x = row0[i0 + m] - cm[i0 + m];
    av.y = row1[i0 + m] - cm[i0 + m];
    bv.x = row0[j0 + m] - cm[j0 + m];
    bv.y = row1[j0 + m] - cm[j0 + m];
    acc = __builtin_amdgcn_wmma_f32_16x16x4_f32(false, av, false, bv,
                                                (short)0, acc, false, false);
  }
#pragma unroll
  for (int r = 0; r < 8; ++r) red[(w * 8 + r) * 32 + lane] = acc[r];
  __syncthreads();
  if (tid < 32) {
    int mhi = (lane >> 4) << 3;
#pragma unroll
    for (int r = 0; r < 8; ++r) {
      float s = 0.f;
#pragma unroll
      for (int ww = 0; ww < 8; ++ww) s += red[(ww * 8 + r) * 32 + lane];
      int M = r + mhi, N = lane & 15;
      ws[WS_COV + (i0 + M) * XCOLS + (j0 + N)] = s / (float)(NROWS - 1);
    }
  }
}

// ---------------------------------------------------------------- IoU count --
__global__ void fnl_iou_kernel(float* __restrict__ ws) {
  __shared__ float svl[RES];
  __shared__ unsigned int cnts[256];
  int tid = threadIdx.x;
  if (tid < RES) svl[tid] = ws[WS_SV + tid];
  __syncthreads();
  int idx = blockIdx.x * blockDim.x + tid;   // 0..65535 = (b,u)
  int b = idx >> 5, u = idx & 31;
  const float* xp = ws + WS_XP + b * XCOLS + u * 3;
  const float* xl = ws + WS_XL + b * XCOLS + u * 3;
  float dx = xp[0] - xl[0], dy = xp[1] - xl[1], dz = xp[2] - xl[2];
  float r = sqrtf(dx * dx + dy * dy + dz * dz);
  unsigned int c = 0;
#pragma unroll
  for (int v = 0; v < RES; ++v) c += (svl[v] * r < 0.5f) ? 1u : 0u;
  cnts[tid] = c;
  __syncthreads();
  for (int s = 128; s > 0; s >>= 1) {
    if (tid < s) cnts[tid] += cnts[tid + s];
    __syncthreads();
  }
  if (tid == 0) atomicAdd((unsigned int*)(ws + WS_CNT), cnts[0]);
}

// -------------------------------------------- finalize: subspace iter + mix --
__global__ void fnl_finalize_kernel(float* __restrict__ ws, float* __restrict__ out) {
  __shared__ float C[XCOLS * XCOLS];    // 36 KB, fits easily in 320 KB LDS
  __shared__ float V[XCOLS][3];
  __shared__ float W[XCOLS][3];
  __shared__ float dvec[XCOLS];
  int tid = threadIdx.x;
  for (int i = tid; i < XCOLS * XCOLS; i += blockDim.x) C[i] = ws[WS_COV + i];
  if (tid < XCOLS) {
    dvec[tid] = ws[WS_MEANP + tid] - ws[WS_MEANL + tid];
    V[tid][0] = sinf(12.9898f * (float)(tid + 1));
    V[tid][1] = sinf(78.2330f * (float)(tid + 1));
    V[tid][2] = sinf(37.7190f * (float)(tid + 1));
  }
  __syncthreads();
  for (int it = 0; it < 100; ++it) {
    if (tid < XCOLS) {
      float w0 = 0.f, w1 = 0.f, w2 = 0.f;
      for (int j = 0; j < XCOLS; ++j) {
        float cij = C[tid * XCOLS + j];
        w0 += cij * V[j][0];
        w1 += cij * V[j][1];
        w2 += cij * V[j][2];
      }
      W[tid][0] = w0; W[tid][1] = w1; W[tid][2] = w2;
    }
    __syncthreads();
    if (tid == 0) {                     // Gram-Schmidt (96x3: trivially cheap)
      for (int t = 0; t < 3; ++t) {
        for (int s = 0; s < t; ++s) {
          float dot = 0.f;
          for (int i = 0; i < XCOLS; ++i) dot += V[i][s] * W[i][t];
          for (int i = 0; i < XCOLS; ++i) W[i][t] -= dot * V[i][s];
        }
        float nrm = 0.f;
        for (int i = 0; i < XCOLS; ++i) nrm += W[i][t] * W[i][t];
        nrm = sqrtf(nrm);
        float inv = (nrm > 1e-20f) ? 1.f / nrm : 0.f;
        for (int i = 0; i < XCOLS; ++i) V[i][t] = W[i][t] * inv;
      }
    }
    __syncthreads();
  }
  if (tid == 0) {
    float p0 = 0.f, p1 = 0.f, p2 = 0.f;
    for (int i = 0; i < XCOLS; ++i) {
      p0 += dvec[i] * V[i][0];
      p1 += dvec[i] * V[i][1];
      p2 += dvec[i] * V[i][2];
    }
    float sv2 = 0.f;
    for (int v = 0; v < RES; ++v) { float s = ws[WS_SV + v]; sv2 += s * s; }
    float dist = sqrtf(p0 * p0 + p1 * p1 + p2 * p2) * sqrtf(sv2);
    float pca_loss = 1.f - dist;
    float inter = (float)(*((unsigned int*)(ws + WS_CNT)));
    float uni = 2048.f - inter;                    // P + P, as in reference
    float iou_loss = 1.f - inter / uni;
    out[0] = pca_loss + iou_loss;                  // LOSS_WEIGHT = 1.0
  }
}

// -------------------------------------------------------------------- launch --
extern "C" void kernel_launch(void* const* d_in, const int* in_sizes, int n_in,
                              void* d_out, int out_size, void* d_ws, size_t ws_size,
                              hipStream_t stream) {
  const float* pred  = (const float*)d_in[0];
  const float* label = (const float*)d_in[1];
  // d_in[2] (target_weight) is unused by the reference.
  float* ws  = (float*)d_ws;
  float* out = (float*)d_out;

  hipLaunchKernelGGL(fnl_basis_kernel,      dim3(1),   dim3(32),  0, stream, ws);
  hipLaunchKernelGGL(fnl_curve_wmma_kernel, dim3(192), dim3(256), 0, stream, pred, label, ws);
  hipLaunchKernelGGL(fnl_means_kernel,      dim3(1),   dim3(256), 0, stream, ws);
  hipLaunchKernelGGL(fnl_cov_wmma_kernel,   dim3(36),  dim3(256), 0, stream, ws);
  hipLaunchKernelGGL(fnl_iou_kernel,        dim3(256), dim3(256), 0, stream, ws);
  hipLaunchKernelGGL(fnl_finalize_kernel,   dim3(1),   dim3(128), 0, stream, ws, out);
}